// DSBatchNorm_1090921693630
// MI455X (gfx1250) — compile-verified
//
#include <hip/hip_runtime.h>
#include <hip/hip_bf16.h>

typedef __attribute__((ext_vector_type(2))) float v2f;
typedef __attribute__((ext_vector_type(8))) float v8f;

#define EPSV 1e-5f
#define DPAD 16   // domain rows padded to 16 for the WMMA M dimension (real D = 8)

// ---------------- scratch zeroing (ws is poisoned 0xAA before timing) ----------------
__global__ void zero_ws_kernel(float* __restrict__ ws, int n) {
  int i = blockIdx.x * blockDim.x + threadIdx.x;
  if (i < n) ws[i] = 0.0f;
}

// ---------------- per-domain counts: LDS reduce then few global atomics -------------
__global__ void count_kernel(const int* __restrict__ y, float* __restrict__ cnt, int n) {
  __shared__ float lc[DPAD];
  int t = threadIdx.x;
  if (t < DPAD) lc[t] = 0.0f;
  __syncthreads();
  int i = blockIdx.x * blockDim.x + t;
  if (i < n) atomicAdd(&lc[y[i] & (DPAD - 1)], 1.0f);
  __syncthreads();
  if (t < DPAD) {
    float v = lc[t];
    if (v != 0.0f) atomicAdd(&cnt[t], v);
  }
}

// ---------------- segment sums as one-hot GEMM on V_WMMA_F32_16X16X4_F32 ------------
// S1[d,f] = sum_n 1{y[n]==d} x[n,f],  S2[d,f] = sum_n 1{y[n]==d} x[n,f]^2
// A (16x4) = one-hot of 4 rows' domains; B (4x16) = x tile (and its square).
// A layout: lane L -> M=L&15; VGPR0 holds K0 (lanes 0-15)/K2 (lanes 16-31), VGPR1 K1/K3.
// B layout: VGPR0 rows K0/K2, VGPR1 rows K1/K3, column = f_base + (L&15).
// C/D layout: VGPR i -> domain i (lanes 0-15) / domain i+8 (lanes 16-31).
// NF = compile-time row stride (0 => use runtime nfeat), so the second-row load folds
// into the 24-bit instruction offset and the loop carries a single pointer increment.
template <int NF>
__global__ void stats_kernel(const float* __restrict__ x, const int* __restrict__ y,
                             float* __restrict__ s1, float* __restrict__ s2,
                             int nrows, int nfeat, int rowchunks) {
  const int stride = NF ? NF : nfeat;
  const int lane = threadIdx.x & 31;
  const int wave = threadIdx.x >> 5;
  const int fbase = blockIdx.x * 16;
  const int chunk = blockIdx.y * (blockDim.x >> 5) + wave;
  if (chunk >= rowchunks) return;          // uniform per wave: EXEC stays all-ones
  const int rows_per_chunk = nrows / rowchunks;   // multiple of 8 by construction
  const int r0 = chunk * rows_per_chunk;

  const int d  = lane & 15;   // M (domain row of A / of output)
  const int hi = lane >> 4;   // 0 -> K0/K1 half, 1 -> K2/K3 half
  const int f  = fbase + d;   // output feature column for this lane

  const int*   py = y + r0 + 2 * hi;
  const float* px = x + (size_t)(r0 + 2 * hi) * stride + f;

  v8f acc1 = {};
  v8f acc2 = {};
  const int iters = rows_per_chunk / 8;    // unroll x2: 8 rows per iteration
  for (int it = 0; it < iters; ++it) {
    // rows r .. r+3
    const int   ya0 = py[0];
    const int   yb0 = py[1];
    const float bx0 = px[0];
    const float by0 = px[stride];
    // rows r+4 .. r+7
    const int   ya1 = py[4];
    const int   yb1 = py[5];
    const float bx1 = px[4 * stride];
    const float by1 = px[5 * stride];

    v2f A0, B10, B20, A1, B11, B21;
    A0.x = (ya0 == d) ? 1.0f : 0.0f;
    A0.y = (yb0 == d) ? 1.0f : 0.0f;
    B10.x = bx0;       B10.y = by0;
    B20.x = bx0 * bx0; B20.y = by0 * by0;
    A1.x = (ya1 == d) ? 1.0f : 0.0f;
    A1.y = (yb1 == d) ? 1.0f : 0.0f;
    B11.x = bx1;       B11.y = by1;
    B21.x = bx1 * bx1; B21.y = by1 * by1;

    acc1 = __builtin_amdgcn_wmma_f32_16x16x4_f32(false, A0, false, B10, (short)0, acc1, false, false);
    acc2 = __builtin_amdgcn_wmma_f32_16x16x4_f32(false, A0, false, B20, (short)0, acc2, false, false);
    acc1 = __builtin_amdgcn_wmma_f32_16x16x4_f32(false, A1, false, B11, (short)0, acc1, false, false);
    acc2 = __builtin_amdgcn_wmma_f32_16x16x4_f32(false, A1, false, B21, (short)0, acc2, false, false);

    py += 8;
    px += (size_t)8 * stride;
  }

  const int drow = hi * 8;   // lanes 16-31 carry (always-zero) domains 8..15
#pragma unroll
  for (int i = 0; i < 8; ++i) {
    atomicAdd(&s1[(size_t)(drow + i) * stride + f], acc1[i]);
    atomicAdd(&s2[(size_t)(drow + i) * stride + f], acc2[i]);
  }
}

// ---------------- mean / inv-std per (d, f) -----------------------------------------
__global__ void finalize_kernel(const float* __restrict__ s1, const float* __restrict__ s2,
                                const float* __restrict__ cnt,
                                float* __restrict__ mean, float* __restrict__ inv,
                                int nfeat, int total) {
  int i = blockIdx.x * blockDim.x + threadIdx.x;
  if (i >= total) return;
  int dd = i / nfeat;
  float c  = cnt[dd];
  float cs = fmaxf(c, 1.0f);
  float m  = s1[i] / cs;
  float v  = s2[i] / cs - m * m;
  mean[i] = m;
  inv[i]  = rsqrtf(v + EPSV);
}

// ---------------- streaming normalize + affine: one block == one row ----------------
// grid = N rows, block = F/4 threads; y[n] and cnt[d] are block-uniform.
__global__ void apply_kernel(const float4* __restrict__ x4, const int* __restrict__ y,
                             const float4* __restrict__ g4, const float4* __restrict__ b4,
                             const float4* __restrict__ m4, const float4* __restrict__ i4,
                             const float* __restrict__ cnt,
                             float4* __restrict__ out4, int f4) {
  const int n  = blockIdx.x;
  const int fi = threadIdx.x;
  const size_t idx = (size_t)n * f4 + fi;
  const int dd = y[n] & 7;
  const float c = cnt[dd];
  const float4 xv = x4[idx];
  float4 o;
  if (c > 1.0f) {
    const size_t p = (size_t)dd * f4 + fi;
    const float4 m  = m4[p];
    const float4 iv = i4[p];
    const float4 g  = g4[p];
    const float4 b  = b4[p];
    o.x = fmaf(g.x, (xv.x - m.x) * iv.x, b.x);
    o.y = fmaf(g.y, (xv.y - m.y) * iv.y, b.y);
    o.z = fmaf(g.z, (xv.z - m.z) * iv.z, b.z);
    o.w = fmaf(g.w, (xv.w - m.w) * iv.w, b.w);
  } else if (c == 1.0f) {
    o = xv;
  } else {
    o = make_float4(0.0f, 0.0f, 0.0f, 0.0f);
  }
  out4[idx] = o;
}

extern "C" void kernel_launch(void* const* d_in, const int* in_sizes, int n_in,
                              void* d_out, int out_size, void* d_ws, size_t ws_size,
                              hipStream_t stream) {
  const float* x     = (const float*)d_in[0];
  const int*   y     = (const int*)d_in[1];
  const float* gamma = (const float*)d_in[2];
  const float* beta  = (const float*)d_in[3];
  float*       out   = (float*)d_out;

  const int n = in_sizes[1];            // 16384 rows
  const int f = in_sizes[0] / n;        // 1024 features

  // ws layout (floats): S1[16*f] | S2[16*f] | cnt[16] | mean[8*f] | inv[8*f]
  float* ws   = (float*)d_ws;
  float* s1   = ws;
  float* s2   = ws + (size_t)DPAD * f;
  float* cnt  = ws + (size_t)2 * DPAD * f;
  float* mean = cnt + DPAD;
  float* inv  = mean + (size_t)8 * f;

  // 1) zero accumulators (S1, S2, cnt)
  const int zn = 2 * DPAD * f + DPAD;
  zero_ws_kernel<<<(zn + 255) / 256, 256, 0, stream>>>(ws, zn);

  // 2) per-domain counts
  count_kernel<<<(n + 255) / 256, 256, 0, stream>>>(y, cnt, n);

  // 3) one-hot-GEMM segment sums via f32 WMMA
  const int rowchunks = 128;                       // 128 rows per wave -> 8192 waves
  dim3 sgrid(f / 16, rowchunks / 8);               // 8 waves per 256-thread block
  if (f == 1024) {
    stats_kernel<1024><<<sgrid, 256, 0, stream>>>(x, y, s1, s2, n, f, rowchunks);
  } else {
    stats_kernel<0><<<sgrid, 256, 0, stream>>>(x, y, s1, s2, n, f, rowchunks);
  }

  // 4) mean / inv-std
  const int ft = 8 * f;
  finalize_kernel<<<(ft + 255) / 256, 256, 0, stream>>>(s1, s2, cnt, mean, inv, f, ft);

  // 5) streaming apply (x should still be L2-resident: 64 MiB < 192 MiB L2)
  apply_kernel<<<n, f / 4, 0, stream>>>(
      (const float4*)x, y, (const float4*)gamma, (const float4*)beta,
      (const float4*)mean, (const float4*)inv, cnt, (float4*)out, f / 4);
}